// CannyDetector_32985348833430
// MI455X (gfx1250) — compile-verified
//
#include <hip/hip_runtime.h>
#include <stdint.h>

// -----------------------------------------------------------------------------
// Canny-style directional edge detector, MI455X (gfx1250).
//
// Roofline: 403 MB traffic, ~0.8 GFLOP -> ~17 us floor at 23.3 TB/s. Pure
// data-movement problem; WMMA is deliberately unused (0.002 FLOP/byte).
// Path: Tensor Data Mover DMAs strided 2D row-tiles into LDS (double-buffered,
// overlapped via TENSORcnt), ds_load_b128 window reads, non-temporal
// global_store_b128 writes. TDM issue is gated by a readfirstlane scalar
// branch (TDM ignores EXEC, so wave-granularity gating is mandatory).
// -----------------------------------------------------------------------------

#define HH 1024
#define WW 1024
#define TILE_ROWS 16           // output rows per step
#define BUF_ROWS  18           // + top/bottom halo
#define SEG       64           // rows per workgroup segment
#define THREADS   256          // 8 wave32s
#define THRESH    (15.0f / 255.0f)

typedef uint32_t u32;
typedef uint64_t u64;
typedef __attribute__((ext_vector_type(4))) float f4;
typedef __attribute__((ext_vector_type(4))) u32   v4u;
typedef __attribute__((ext_vector_type(4))) int   v4i;
typedef __attribute__((ext_vector_type(8))) int   v8i;

// Require the CDNA5 TDM path: if this fails to compile, the toolchain lacks
// the builtins and we need a different strategy -- no silent fallback.
#if defined(__HIP_DEVICE_COMPILE__)
#if !__has_builtin(__builtin_amdgcn_tensor_load_to_lds) || \
    !__has_builtin(__builtin_amdgcn_s_wait_tensorcnt)
#error "gfx1250 TDM builtins unavailable on this toolchain"
#endif
#endif

struct RowV { float l; f4 m; float r; };

// Load the x4..x4+3 window of image row r_img (plus left/right halo scalars)
// from the LDS tile; rows/cols outside the image read as zero (= conv padding).
// Row-validity condition is wave-uniform (scalar branch); column edges are
// lane-divergent v_cndmask selects.
__device__ __forceinline__ RowV load_row(const float* __restrict__ buf,
                                         int r_img, int gy0, int h, int x4) {
    RowV o;
    if (r_img < gy0 || r_img >= gy0 + h) {
        o.l = 0.0f; o.r = 0.0f;
        o.m = (f4){0.0f, 0.0f, 0.0f, 0.0f};
        return o;
    }
    const float* row = buf + (size_t)(r_img - gy0) * WW;
    o.m = *(const f4*)(row + x4);            // ds_load_b128 (16B aligned)
    o.l = (x4 == 0)      ? 0.0f : row[x4 - 1];
    o.r = (x4 == WW - 4) ? 0.0f : row[x4 + 4];
    return o;
}

// 4 directional |diff| >= T, OR'd, for 4 consecutive pixels. A=y-1, B=y, C=y+1.
__device__ __forceinline__ f4 stencil(const RowV& A, const RowV& B, const RowV& C) {
    float Am[4] = {A.m.x, A.m.y, A.m.z, A.m.w};
    float Cm[4] = {C.m.x, C.m.y, C.m.z, C.m.w};
    float Al[4] = {A.l,   A.m.x, A.m.y, A.m.z};
    float Ar[4] = {A.m.y, A.m.z, A.m.w, A.r  };
    float Bl[4] = {B.l,   B.m.x, B.m.y, B.m.z};
    float Br[4] = {B.m.y, B.m.z, B.m.w, B.r  };
    float Cl[4] = {C.l,   C.m.x, C.m.y, C.m.z};
    float Cr[4] = {C.m.y, C.m.z, C.m.w, C.r  };
    float out[4];
#pragma unroll
    for (int i = 0; i < 4; ++i) {
        bool e = (fabsf(Bl[i] - Br[i]) >= THRESH)    // 0 deg
              || (fabsf(Al[i] - Cr[i]) >= THRESH)    // 45 deg
              || (fabsf(Am[i] - Cm[i]) >= THRESH)    // 90 deg
              || (fabsf(Ar[i] - Cl[i]) >= THRESH);   // 135 deg
        out[i] = e ? 1.0f : 0.0f;
    }
    f4 o; o.x = out[0]; o.y = out[1]; o.z = out[2]; o.w = out[3];
    return o;
}

// Issue one TDM 2D tile load: h rows x 1024 f32, row stride 1024 elements,
// into LDS at byte offset lds_byte_off. D# layout per CDNA5 ISA ch.8:
//   group0: count=1 | lds_addr | global_addr[56:0] | type=2
//   group1: data_size=4B | tensor_dim0/1 | tile_dim0/1 | tensor_dim0_stride
//   groups 2/3: zero (2D tile, no iterate/gather/multicast)
__device__ __forceinline__ void tdm_issue(u32 lds_byte_off,
                                          const float* gsrc, int h) {
#if defined(__HIP_DEVICE_COMPILE__)
    u64 ga = (u64)(uintptr_t)gsrc;
    v4u g0;
    g0[0] = 1u;                                        // count=1 (valid), user mode
    g0[1] = lds_byte_off;                              // lds_addr (bytes)
    g0[2] = (u32)ga;                                   // global_addr[31:0]
    g0[3] = ((u32)(ga >> 32) & 0x01FFFFFFu)            // global_addr[56:32]
          | 0x80000000u;                               // type=2 ("image")
    v8i g1;
    g1[0] = 0x00020000;                                // data_size=2 (4 bytes)
    g1[1] = (int)((u32)WW << 16);                      // tensor_dim0 lo16 = 1024
    g1[2] = (int)((u32)h  << 16);                      // dim0 hi=0 | tensor_dim1 lo = h
    g1[3] = (int)((u32)WW << 16);                      // dim1 hi=0 | tile_dim0 = 1024
    g1[4] = h;                                         // tile_dim1 = h, tile_dim2 = 0
    g1[5] = WW;                                        // tensor_dim0_stride lo32 = 1024
    g1[6] = 0;                                         // stride hi | dim1_stride lo
    g1[7] = 0;
    v4i z4 = {0, 0, 0, 0};                             // groups 2/3 unused (2D tile)
#if defined(__clang_major__) && __clang_major__ >= 23
    v8i z8 = {0, 0, 0, 0, 0, 0, 0, 0};
    __builtin_amdgcn_tensor_load_to_lds(g0, g1, z4, z4, z8, 0);
#else
    __builtin_amdgcn_tensor_load_to_lds(g0, g1, z4, z4, 0);
#endif
#else
    (void)lds_byte_off; (void)gsrc; (void)h;
#endif
}

__device__ __forceinline__ void wait_tensorcnt_le1() {
#if defined(__HIP_DEVICE_COMPILE__)
    __builtin_amdgcn_s_wait_tensorcnt(1);
#endif
}
__device__ __forceinline__ void wait_tensorcnt_0() {
#if defined(__HIP_DEVICE_COMPILE__)
    __builtin_amdgcn_s_wait_tensorcnt(0);
#endif
}

__global__ __launch_bounds__(THREADS)
void canny_edges_kernel(const float* __restrict__ in, float* __restrict__ out) {
    extern __shared__ float lds[];                    // 2 x BUF_ROWS x WW floats
    const int plane = blockIdx.y;
    const int seg   = blockIdx.x;
    const float* src = in  + (size_t)plane * HH * WW;
    float*       dst = out + (size_t)plane * HH * WW;
    const int ybase = seg * SEG;
    const int steps = SEG / TILE_ROWS;
    const int tid   = threadIdx.x;
    const int x4    = tid * 4;                        // one float4 column / thread

    // TDM ignores EXEC -> gate at wave granularity with a *scalar* branch.
    // readfirstlane(tid) == wave_id*32, so this is true only for wave 0 and
    // lowers to s_cmp + s_cbranch (verified in round-1 asm).
    const bool issuer = (__builtin_amdgcn_readfirstlane(tid) == 0);

    {   // Prologue: DMA tile 0 (top halo clipped at image edge).
        int y0 = ybase;
        int lo = (y0 - 1 < 0) ? 0 : y0 - 1;
        int hi = (y0 + TILE_ROWS + 1 > HH) ? HH : y0 + TILE_ROWS + 1;
        if (issuer) tdm_issue(0u, src + (size_t)lo * WW, hi - lo);
    }

    for (int s = 0; s < steps; ++s) {
        const int y0  = ybase + s * TILE_ROWS;
        const int gy0 = (y0 - 1 < 0) ? 0 : y0 - 1;
        const int ghi = (y0 + TILE_ROWS + 1 > HH) ? HH : y0 + TILE_ROWS + 1;
        const int h   = ghi - gy0;
        float* bufp = lds + (size_t)(s & 1) * BUF_ROWS * WW;

        if (issuer) {
            if (s + 1 < steps) {
                // Prefetch next tile into the other buffer, then wait for the
                // current one (TDM ops from one wave complete in order, so
                // TENSORcnt<=1 means tile s has landed in LDS).
                int ny0 = ybase + (s + 1) * TILE_ROWS;
                int nlo = ny0 - 1;                    // interior: never clipped
                int nhi = (ny0 + TILE_ROWS + 1 > HH) ? HH : ny0 + TILE_ROWS + 1;
                tdm_issue((u32)(((s + 1) & 1) * BUF_ROWS * WW * sizeof(float)),
                          src + (size_t)nlo * WW, nhi - nlo);
                wait_tensorcnt_le1();
            } else {
                wait_tensorcnt_0();
            }
        }
        __syncthreads();                              // publish LDS tile to all waves

        // Rolling 3-row window: 18 LDS row reads for 16 output rows
        // (1x ds_load_b128 + 2x ds_load_b32 per new row).
        RowV A = load_row(bufp, y0 - 1, gy0, h, x4);
        RowV B = load_row(bufp, y0,     gy0, h, x4);
#pragma unroll 4
        for (int r = 0; r < TILE_ROWS; ++r) {
            RowV C = load_row(bufp, y0 + r + 1, gy0, h, x4);
            f4 e = stencil(A, B, C);
            // write-once output: non-temporal 128-bit store, keep L2 for input
            __builtin_nontemporal_store(e, (f4*)(dst + (size_t)(y0 + r) * WW + x4));
            A = B; B = C;
        }
        __syncthreads();           // buffer s&1 reusable when step s+2 DMAs into it
    }
}

extern "C" void kernel_launch(void* const* d_in, const int* in_sizes, int n_in,
                              void* d_out, int out_size, void* d_ws, size_t ws_size,
                              hipStream_t stream) {
    (void)n_in; (void)d_ws; (void)ws_size; (void)out_size;
    const float* imgs = (const float*)d_in[0];
    float* edges = (float*)d_out;
    const int planes = in_sizes[0] / (HH * WW);       // B*C = 48
    dim3 grid(HH / SEG, planes);                      // 16 x 48 = 768 workgroups
    dim3 block(THREADS);
    size_t shmem = (size_t)2 * BUF_ROWS * WW * sizeof(float); // 144 KB (< 320 KB/WGP)
    canny_edges_kernel<<<grid, block, shmem, stream>>>(imgs, edges);
}